// QuantumRecurrentUnit_51015621542292
// MI455X (gfx1250) — compile-verified
//
#include <hip/hip_runtime.h>
#include <math.h>

typedef __attribute__((ext_vector_type(2))) float v2f;
typedef __attribute__((ext_vector_type(8))) float v8f;
typedef __attribute__((ext_vector_type(4))) unsigned int v4u;
typedef __attribute__((ext_vector_type(8))) int v8i;
typedef __attribute__((ext_vector_type(4))) int v4i;

// ---------------------------------------------------------------------------
// Kernel 1: build G[s][t] = W[t][P^{-1} s], W = kron of RY(w_k), P = CNOT ring.
// Qubit k lives at bit (7-k). Stored K-pair interleaved:
//   ws[(s>>1)*512 + t*2 + (s&1)]  -> a WMMA B-fragment is one b64 load/lane,
//   and a 16-column panel is a clean 2D tile (128 rows x 128B, 2KB stride)
//   for the Tensor Data Mover.
// ---------------------------------------------------------------------------
__global__ void qru_build_G(const float* __restrict__ w, float* __restrict__ G) {
    const int t = threadIdx.x;              // 0..255
    float c[8], s[8];
#pragma unroll
    for (int k = 0; k < 8; ++k) {
        float a = 0.5f * w[k];
        c[k] = cosf(a);
        s[k] = sinf(a);
    }
    for (int sIdx = 0; sIdx < 256; ++sIdx) {
        // u = P^{-1}(s): CNOT(ctrl=qubit i, tgt=qubit (i+1)%8) xor-maps in
        // forward gate order i=0..7 (ctrl bit 7-i, tgt bit (6-i)&7).
        unsigned u = (unsigned)sIdx;
#pragma unroll
        for (int i = 0; i < 8; ++i) {
            const int cb = 7 - i;
            const int tb = (6 - i) & 7;
            u ^= ((u >> cb) & 1u) << tb;
        }
        // W[t][u] = prod_k M_k[t_k][u_k], M = [[c,-s],[s,c]]
        float val = 1.0f;
#pragma unroll
        for (int k = 0; k < 8; ++k) {
            const int a = (t >> (7 - k)) & 1;
            const int b = (u >> (7 - k)) & 1;
            const float e = (a == b) ? c[k] : (a ? s[k] : -s[k]);
            val *= e;
        }
        G[((size_t)(sIdx >> 1) * 256 + t) * 2 + (sIdx & 1)] = val;
    }
}

// ---------------------------------------------------------------------------
// TDM: stage one 16-column G panel (128 rows x 32 floats, row stride 512
// floats) into LDS.  D# per CDNA5 ISA §8.3/8.4: group0 {count=1, lds_addr,
// global_addr, type=2}, group1 {data_size=4B, tensor 512x128, tile 32x128,
// dim0_stride=512}.  Groups 2/3 zero (2D tensor).  This toolchain's builtin
// takes 6 args: (g0, g1, g2, g3, extra, cpol).
// ---------------------------------------------------------------------------
__device__ __forceinline__ void qru_tdm_load_panel(const float* gsrc,
                                                   unsigned int ldsOff) {
    const unsigned long long ga = (unsigned long long)(uintptr_t)gsrc;
    const v4u g0 = {1u,                                   // count=1, user mode
                    ldsOff,                               // lds_addr
                    (unsigned int)ga,                     // global_addr[31:0]
                    (unsigned int)((ga >> 32) & 0x1FFFFFFu) | (2u << 30)}; // type=2
    const v8i g1 = {(int)(2u << 16),     // data_size = 2 (4 bytes)
                    (int)(512u << 16),   // tensor_dim0 = 512 (lo16)
                    (int)(128u << 16),   // tensor_dim0 hi=0 | tensor_dim1 = 128
                    (int)(32u << 16),    // tensor_dim1 hi=0 | tile_dim0 = 32
                    (int)128,            // tile_dim1 = 128, tile_dim2 = 0
                    (int)512,            // tensor_dim0_stride = 512
                    0, 0};
    const v4i z4 = {0, 0, 0, 0};
    const v8i z8 = {0, 0, 0, 0, 0, 0, 0, 0};
    __builtin_amdgcn_tensor_load_to_lds(g0, g1, z4, z4, z8, 0);
}

// ---------------------------------------------------------------------------
// Kernel 2: fused  V build -> Y = V * G (WMMA f32 16x16x4) -> z = (Y^2)*signs
// 128 threads = 4 waves; 16 samples/wave; G panels TDM-staged, double-buffered.
// Dynamic LDS: 64KB V (16x256/wave, swizzled) + 2 x 16KB G panels = 96KB.
// ---------------------------------------------------------------------------
#define QRU_WAVES 4
#define QRU_V_FLOATS (QRU_WAVES * 16 * 256)   // 16384 floats = 64KB
#define QRU_PANEL_FLOATS (128 * 32)           // 4096 floats = 16KB

__global__ __launch_bounds__(128) void qru_gemm(const float* __restrict__ x,
                                                const float* __restrict__ G,
                                                float* __restrict__ out) {
    extern __shared__ float smem[];
    float* Vlds = smem;                          // [QRU_V_FLOATS]
    float* Gp   = smem + QRU_V_FLOATS;           // [2][QRU_PANEL_FLOATS]

    const int tid  = threadIdx.x;
    const int wave = tid >> 5;
    const int lane = tid & 31;
    const int half = lane >> 4;      // WMMA lane-half: selects K pair +0 / +2
    const int m    = lane & 15;      // row (sample) in tile / N within col-tile
    const int sw   = 4 * m;          // LDS bank swizzle (multiple of 4)

    const int waveBase = blockIdx.x * (QRU_WAVES * 16) + wave * 16;

    // Kick off TDM for panel 0 before the (long-latency) trig work.
    if (wave == 0)
        qru_tdm_load_panel(G, (unsigned int)(uintptr_t)&Gp[0]);

    // ---- Phase 1: product state V[m][s] = prod_i f_i(bit_{7-i}(s)) ----------
    {
        const float* xa = x + (size_t)(waveBase + m) * 16;
        float c[8], s[8];
#pragma unroll
        for (int k = 0; k < 8; ++k) {
            float a = 0.5f * xa[k];
            c[k] = cosf(a);
            s[k] = sinf(a);
        }
        // hi nibble (bits 7..4) = qubits 0..3; bit7 == half for this s-range.
        const float f0 = half ? s[0] : c[0];
        float t12[4] = {c[1] * c[2], c[1] * s[2], s[1] * c[2], s[1] * s[2]};
        float ph[8];
#pragma unroll
        for (int i = 0; i < 8; ++i)
            ph[i] = f0 * t12[i >> 1] * ((i & 1) ? s[3] : c[3]);
        // lo nibble (bits 3..0) = qubits 4..7
        float t45[4] = {c[4] * c[5], c[4] * s[5], s[4] * c[5], s[4] * s[5]};
        float t67[4] = {c[6] * c[7], c[6] * s[7], s[6] * c[7], s[6] * s[7]};
        float pl[16];
#pragma unroll
        for (int l = 0; l < 16; ++l)
            pl[l] = t45[l >> 2] * t67[l & 3];

        float* Vw = &Vlds[wave * 16 * 256];
#pragma unroll
        for (int jh = 0; jh < 8; ++jh) {
            const int kbase = (half * 8 + jh) * 16;
#pragma unroll
            for (int lo = 0; lo < 16; lo += 2) {
                v2f pr = {ph[jh] * pl[lo], ph[jh] * pl[lo + 1]};
                *reinterpret_cast<v2f*>(&Vw[m * 256 + ((kbase + lo) ^ sw)]) = pr;
            }
        }
    }

    // ---- Phase 2: WMMA GEMM, TDM double-buffered B, fused epilogue ----------
    const float* Vw = &Vlds[wave * 16 * 256];
    float psum[8];                       // sum_t p   (k>=4: lane-bit signs)
    float z03[4][8];                     // sum_t p*sign_k (k<=3: ntile signs)
#pragma unroll
    for (int r = 0; r < 8; ++r) {
        psum[r] = 0.f;
#pragma unroll
        for (int k = 0; k < 4; ++k) z03[k][r] = 0.f;
    }

    const v8f zero8 = {0.f, 0.f, 0.f, 0.f, 0.f, 0.f, 0.f, 0.f};

    for (int nt = 0; nt < 16; ++nt) {
        if (wave == 0)
            __builtin_amdgcn_s_wait_tensorcnt(0);   // panel nt landed in LDS
        __syncthreads();  // panel nt visible; everyone done reading other buf
        if (wave == 0 && nt < 15)   // prefetch next panel into the other buffer
            qru_tdm_load_panel(G + (nt + 1) * 32,
                               (unsigned int)(uintptr_t)&Gp[((nt + 1) & 1) *
                                                            QRU_PANEL_FLOATS]);

        const float* Pb = &Gp[(nt & 1) * QRU_PANEL_FLOATS];

        v8f acc0 = zero8, acc1 = zero8;  // two chains break D->C dependency
#pragma unroll 4
        for (int kk = 0; kk < 64; kk += 2) {
            // A fragment (ISA layout): lanes = M, VGPR0/1 = K {4kk,4kk+1},
            // lane-half selects +2.  Swizzled LDS b64 load, conflict-free.
            v2f a0 = *reinterpret_cast<const v2f*>(
                &Vw[m * 256 + ((4 * kk + 2 * half) ^ sw)]);
            v2f a1 = *reinterpret_cast<const v2f*>(
                &Vw[m * 256 + ((4 * (kk + 1) + 2 * half) ^ sw)]);
            // B fragment from TDM-staged panel: row j = 2kk+half (32 floats),
            // element pair at 2m.  Banks {2m,2m+1} / {2m+32,+33}: conflict-free.
            v2f b0 = *reinterpret_cast<const v2f*>(
                &Pb[(2 * kk + half) * 32 + 2 * m]);
            v2f b1 = *reinterpret_cast<const v2f*>(
                &Pb[(2 * (kk + 1) + half) * 32 + 2 * m]);
            acc0 = __builtin_amdgcn_wmma_f32_16x16x4_f32(
                false, a0, false, b0, (short)0, acc0, false, false);
            acc1 = __builtin_amdgcn_wmma_f32_16x16x4_f32(
                false, a1, false, b1, (short)0, acc1, false, false);
        }

        // Epilogue: p = Y^2; k<=3 sign depends only on nt bits (wave-uniform)
#pragma unroll
        for (int r = 0; r < 8; ++r) {
            const float y = acc0[r] + acc1[r];
            const float p = y * y;
            psum[r] += p;
#pragma unroll
            for (int k = 0; k < 4; ++k)
                z03[k][r] += ((nt >> (3 - k)) & 1) ? -p : p;
        }
    }

    // ---- Final: cross-lane reduce over the 16 t-columns of each half --------
#pragma unroll
    for (int r = 0; r < 8; ++r) {
        const int sampleOut = waveBase + r + 8 * half;  // D layout: M=r / r+8
#pragma unroll
        for (int k = 0; k < 8; ++k) {
            float v;
            if (k < 4) {
                v = z03[k][r];
            } else {
                // sign_k(t) from bit (7-k) of t's low nibble (= lane bits)
                const int bit = (lane >> (7 - k)) & 1;
                v = bit ? -psum[r] : psum[r];
            }
            v += __shfl_xor(v, 1);
            v += __shfl_xor(v, 2);
            v += __shfl_xor(v, 4);
            v += __shfl_xor(v, 8);
            if (m == k) out[(size_t)sampleOut * 8 + k] = v;
        }
    }
}

// ---------------------------------------------------------------------------
extern "C" void kernel_launch(void* const* d_in, const int* in_sizes, int n_in,
                              void* d_out, int out_size, void* d_ws, size_t ws_size,
                              hipStream_t stream) {
    const float* x = (const float*)d_in[0];   // (128,256,16) f32
    const float* w = (const float*)d_in[1];   // (8,) f32
    float* out = (float*)d_out;               // (128,256,8) f32
    float* G = (float*)d_ws;                  // 256*256 f32 = 256 KB scratch

    const int samples = in_sizes[0] / 16;     // 32768
    const int blocks = samples / (QRU_WAVES * 16);
    const size_t shmem = (QRU_V_FLOATS + 2 * QRU_PANEL_FLOATS) * sizeof(float);

    qru_build_G<<<1, 256, 0, stream>>>(w, G);
    qru_gemm<<<blocks, 128, shmem, stream>>>(x, G, out);
}